// FeasibilityLoss_21311627722883
// MI455X (gfx1250) — compile-verified
//
#include <hip/hip_runtime.h>
#include <math.h>

// ---------------- problem constants ----------------
#define BATCH 1024
#define TPTS  1024
#define KQ    15
#define KT    20
#define NDOF  7

// ---------------- output layout (flat float offsets, return-tuple order) ----
#define O_MODEL   0LL
#define O_QDL     1LL
#define O_QDDL    (1LL + 7168LL)
#define O_QDDDL   (1LL + 2LL*7168LL)
#define O_TQL     (1LL + 3LL*7168LL)
#define O_Q       (1LL + 4LL*7168LL)            /* 28673 */
#define O_QDOT    (O_Q + 7340032LL)
#define O_QDDOT   (O_QDOT + 7340032LL)
#define O_QDDDOT  (O_QDDOT + 7340032LL)
#define O_TORQUE  (O_QDDDOT + 7340032LL)        /* 29388801 */
#define O_T       (O_TORQUE + 7340032LL)
#define O_TCUM    (O_T + 1024LL)
#define O_DT      (O_TCUM + 1048576LL)
#define BT        (1024LL*1024LL)

typedef __attribute__((ext_vector_type(2))) float v2f;
typedef __attribute__((ext_vector_type(8))) float v8f;

// ---------------- zero-fill helper ----------------
__global__ void zero_range_kernel(float* __restrict__ out, long long off, long long n) {
  long long i = (long long)blockIdx.x * blockDim.x + threadIdx.x;
  const long long stride = (long long)gridDim.x * blockDim.x;
  for (; i < n; i += stride) out[off + i] = 0.0f;
}

// ---------------- kernel 1: time spline + cumsum (one block per batch) -----
// Writes dtau_dt / ddtau_dtt / dddtau_dttt into the torque region (scratch,
// re-zeroed after kernel 2), plus dt, t_cumsum, t into their final slots.
__global__ __launch_bounds__(256)
void time_spline_kernel(const float* __restrict__ t_cps,
                        const float* __restrict__ Nt,
                        const float* __restrict__ dNt,
                        const float* __restrict__ ddNt,
                        float* __restrict__ out) {
  const int b = blockIdx.x;
  const int tid = threadIdx.x;
  __shared__ float tc[KT];
  __shared__ float part[256];
  if (tid < KT) tc[tid] = t_cps[b * KT + tid];
  __syncthreads();

  float* dtau  = out + O_TORQUE;
  float* ddtau = out + O_TORQUE + BT;
  float* d3tau = out + O_TORQUE + 2 * BT;

  float dloc[4];
  float run = 0.0f;
#pragma unroll
  for (int j = 0; j < 4; ++j) {
    const int t = tid * 4 + j;
    const float* r0 = Nt   + (long long)t * KT;
    const float* r1 = dNt  + (long long)t * KT;
    const float* r2 = ddNt + (long long)t * KT;
    float s0 = 0.f, s1 = 0.f, s2 = 0.f;
#pragma unroll
    for (int k = 0; k < KT; ++k) {
      const float w = tc[k];
      s0 = fmaf(r0[k], w, s0);
      s1 = fmaf(r1[k], w, s1);
      s2 = fmaf(r2[k], w, s2);
    }
    const long long bt = (long long)b * TPTS + t;
    dtau[bt] = s0;  ddtau[bt] = s1;  d3tau[bt] = s2;
    const float d = (1.0f / s0) * (1.0f / (float)TPTS);   // 1/dtau/T
    dloc[j] = d;  run += d;
    out[O_DT + bt] = d;
  }

  // block-wide inclusive scan of per-thread chunk sums (Hillis-Steele)
  part[tid] = run;
  __syncthreads();
  for (int off = 1; off < 256; off <<= 1) {
    const float v = (tid >= off) ? part[tid - off] : 0.0f;
    __syncthreads();
    part[tid] += v;
    __syncthreads();
  }
  float acc = (tid > 0) ? part[tid - 1] : 0.0f;
#pragma unroll
  for (int j = 0; j < 4; ++j) {
    acc += dloc[j];
    out[O_TCUM + (long long)b * TPTS + tid * 4 + j] = acc;
  }
  if (tid == 255) out[O_T + b] = part[255];
}

// ---------------- kernel 2: WMMA spline GEMM + fused epilogue --------------
__device__ __forceinline__ float huber_pen(float x, float lim) {
  float e = fabsf(x) - lim;
  e = fmaxf(e, 0.0f);                 // relu
  return (e < 1.0f) ? (0.5f * e * e) : (e - 0.5f);   // huber, delta = 1
}

// Accumulate one 16(t) x 16(col) tile: C += Nbasis_tile(16x16pad) @ Bfrag
__device__ __forceinline__ v8f mmac(const float* __restrict__ basis, int arow15,
                                    int kb_base, const v2f bfr[4], v8f acc) {
#pragma unroll
  for (int kk = 0; kk < 4; ++kk) {
    const int kb = kk * 4 + kb_base;          // K index of this lane's A pair
    v2f a;
    a.x = (kb     < KQ) ? basis[arow15 + kb]     : 0.0f;
    a.y = (kb + 1 < KQ) ? basis[arow15 + kb + 1] : 0.0f;
    acc = __builtin_amdgcn_wmma_f32_16x16x4_f32(
        /*neg_a=*/false, a, /*neg_b=*/false, bfr[kk],
        /*c_mod=*/(short)0, acc, /*reuse_a=*/false, /*reuse_b=*/false);
  }
  return acc;
}

__global__ __launch_bounds__(32)
void feas_wmma_kernel(const float* __restrict__ q_cps,
                      const float* __restrict__ Nb,  const float* __restrict__ dNb,
                      const float* __restrict__ ddNb, const float* __restrict__ dddNb,
                      const float* __restrict__ qdl, const float* __restrict__ qddl,
                      const float* __restrict__ qdddl,
                      float* __restrict__ out) {
  const int lane  = threadIdx.x;        // one wave32 per block
  const int ct    = blockIdx.x;         // 0..447 column tiles (B*7 = 448*16)
  const int chunk = blockIdx.y;         // 0..7 -> 8 t-tiles each
  const int half  = lane >> 4;
  const int c = ct * 16 + (lane & 15);  // global column = b*7 + m
  const int b = c / NDOF;
  const int m = c - b * NDOF;
  const float l0 = qdl[m], l1 = qddl[m], l2 = qdddl[m];

  // B fragments (4x16 f32 per k-step): lane-half selects K-pair, VGPR within pair
  const int kb_base = half * 2;
  v2f bfr[4];
#pragma unroll
  for (int kk = 0; kk < 4; ++kk) {
    const int k0 = kk * 4 + kb_base;
    bfr[kk].x = (k0     < KQ) ? q_cps[b * (KQ * NDOF) + k0 * NDOF + m]       : 0.0f;
    bfr[kk].y = (k0 + 1 < KQ) ? q_cps[b * (KQ * NDOF) + (k0 + 1) * NDOF + m] : 0.0f;
  }

  const float* dtau_p  = out + O_TORQUE;            // scratch from kernel 1
  const float* ddtau_p = out + O_TORQUE + BT;
  const float* d3tau_p = out + O_TORQUE + 2 * BT;
  const float* dt_p    = out + O_DT;

  float acc0 = 0.f, acc1 = 0.f, acc2 = 0.f;
  const int rm = lane & 15;

  for (int tt = chunk * 8; tt < chunk * 8 + 8; ++tt) {
    const int t0 = tt * 16;
    const int arow15 = (t0 + rm) * KQ;
    v8f cq = {}, cqd = {}, cqdd = {}, cqddd = {};
    cq    = mmac(Nb,    arow15, kb_base, bfr, cq);
    cqd   = mmac(dNb,   arow15, kb_base, bfr, cqd);
    cqdd  = mmac(ddNb,  arow15, kb_base, bfr, cqdd);
    cqddd = mmac(dddNb, arow15, kb_base, bfr, cqddd);

#pragma unroll
    for (int r = 0; r < 8; ++r) {
      const int t = t0 + half * 8 + r;              // C/D layout: M = r + half*8
      const long long bt = (long long)b * TPTS + t;
      const float s   = dtau_p[bt];
      const float sd  = ddtau_p[bt];
      const float sdd = d3tau_p[bt];
      const float dtv = dt_p[bt];
      const float s2  = s * s;

      const float qv    = cq[r];
      const float qdt   = cqd[r];
      const float qddt  = cqdd[r];
      const float qdddt = cqddd[r];

      const float qdot   = qdt * s;
      const float qddot  = qddt * s2 + sd * qdt * s;
      const float qdddot = qdddt * s2 * s + 3.0f * qddt * sd * s2
                         + qdt * s2 * sdd + qdt * sd * sd * s;

      const long long base = (long long)b * (TPTS * NDOF) + (long long)t * NDOF + m;
      out[O_Q      + base] = qv;
      out[O_QDOT   + base] = qdot;
      out[O_QDDOT  + base] = qddot;
      out[O_QDDDOT + base] = qdddot;

      acc0 += huber_pen(qdot,   l0) * dtv;
      acc1 += huber_pen(qddot,  l1) * dtv;
      acc2 += huber_pen(qdddot, l2) * dtv;
    }
  }

  atomicAdd(out + O_QDL   + c, acc0);
  atomicAdd(out + O_QDDL  + c, acc1);
  atomicAdd(out + O_QDDDL + c, acc2);
  atomicAdd(out + O_MODEL, acc0 + acc1 + acc2);
}

// ---------------- launch ----------------
extern "C" void kernel_launch(void* const* d_in, const int* in_sizes, int n_in,
                              void* d_out, int out_size, void* d_ws, size_t ws_size,
                              hipStream_t stream) {
  const float* q_cps = (const float*)d_in[0];
  const float* t_cps = (const float*)d_in[1];
  const float* N     = (const float*)d_in[2];
  const float* dN    = (const float*)d_in[3];
  const float* ddN   = (const float*)d_in[4];
  const float* dddN  = (const float*)d_in[5];
  const float* Nt    = (const float*)d_in[6];
  const float* dNt   = (const float*)d_in[7];
  const float* ddNt  = (const float*)d_in[8];
  const float* qdl   = (const float*)d_in[9];
  const float* qddl  = (const float*)d_in[10];
  const float* qdddl = (const float*)d_in[11];
  (void)in_sizes; (void)n_in; (void)out_size; (void)d_ws; (void)ws_size;
  float* out = (float*)d_out;

  // 1) zero model_loss + 4 loss blocks (atomics accumulate into them)
  zero_range_kernel<<<112, 256, 0, stream>>>(out, 0LL, O_Q);
  // 2) time spline, cumsum; dtau derivatives staged in torque region
  time_spline_kernel<<<BATCH, 256, 0, stream>>>(t_cps, Nt, dNt, ddNt, out);
  // 3) WMMA spline GEMMs + fused derivatives/stores/losses
  dim3 g2(448, 8);
  feas_wmma_kernel<<<g2, 32, 0, stream>>>(q_cps, N, dN, ddN, dddN,
                                          qdl, qddl, qdddl, out);
  // 4) torque output is exactly zero -> wipe the scratch region last
  zero_range_kernel<<<8192, 256, 0, stream>>>(out, O_TORQUE, 7340032LL);
}